// GAT_8787503088148
// MI455X (gfx1250) — compile-verified
//
#include <hip/hip_runtime.h>
#include <math.h>

typedef __attribute__((ext_vector_type(2))) float v2f;
typedef __attribute__((ext_vector_type(8))) float v8f;

#define LEAKY_SLOPE 0.2f
#define BN_EPS_F 1e-5f

__device__ __forceinline__ void atomicMaxF(float* addr, float val) {
    unsigned int* ua = (unsigned int*)addr;
    unsigned int old = __float_as_uint(*addr);
    while (__uint_as_float(old) < val) {
        unsigned int assumed = old;
        old = atomicCAS(ua, assumed, __float_as_uint(val));
        if (old == assumed) break;
    }
}

// H[n, fout] = X[n, fin] @ W[fin, fout], fp32 WMMA 16x16x4.
// Block: (32, fout/16) -> one wave per 16-col tile; 16-row tile staged in LDS.
__global__ void gat_gemm_wmma(const float* __restrict__ X, const float* __restrict__ W,
                              float* __restrict__ H, int n, int fin, int fout) {
    extern __shared__ float lds[];          // 16 x (fin+4), padded vs bank conflicts
    const int ldsStride = fin + 4;
    const int row0 = blockIdx.x * 16;
    const int nThreads = blockDim.x * blockDim.y;
    const int t = threadIdx.y * 32 + threadIdx.x;
    for (int i = t; i < 16 * fin; i += nThreads) {
        int r = i / fin, c = i - r * fin;
        int gr = row0 + r;
        lds[r * ldsStride + c] = (gr < n) ? X[(size_t)gr * fin + c] : 0.0f;
    }
    __syncthreads();

    const int lane = threadIdx.x;           // wave32
    const int m = lane & 15;                // A row / B,D column
    const int kb = (lane >> 4) << 1;        // K pair base: 0 (lanes 0-15) or 2 (lanes 16-31)
    const int colBase = threadIdx.y * 16;

    v8f acc = {};
    for (int k = 0; k < fin; k += 4) {
        v2f a, b;
        // A 16x4 f32: VGPR0 = K={0|2}, VGPR1 = K={1|3}
        a.x = lds[m * ldsStride + k + kb];
        a.y = lds[m * ldsStride + k + kb + 1];
        // B 4x16 f32: same K split, column striped across lanes
        const float* wp = W + (size_t)(k + kb) * fout + colBase + m;
        b.x = wp[0];
        b.y = wp[fout];
        acc = __builtin_amdgcn_wmma_f32_16x16x4_f32(false, a, false, b,
                                                    (short)0, acc, false, false);
    }
    // D 16x16 f32: VGPR v -> row v (lanes 0-15) / row v+8 (lanes 16-31)
    const int rh = (lane >> 4) * 8;
    for (int v = 0; v < 8; ++v) {
        int r = row0 + rh + v;
        if (r < n) H[(size_t)r * fout + colBase + m] = acc[v];
    }
}

// al_src[n,h] = sum_c h[n,h,c]*a_src[h,c];  al_dst likewise
__global__ void att_coef_kernel(const float* __restrict__ Hf,
                                const float* __restrict__ a_src, const float* __restrict__ a_dst,
                                float* __restrict__ als, float* __restrict__ ald,
                                int N, int heads, int C) {
    int t = blockIdx.x * blockDim.x + threadIdx.x;
    if (t >= N * heads) return;
    int node = t / heads, h = t - node * heads;
    const float* hp = Hf + (size_t)node * heads * C + h * C;
    float s1 = 0.0f, s2 = 0.0f;
    for (int c = 0; c < C; ++c) {
        float v = hp[c];
        s1 += v * a_src[h * C + c];
        s2 += v * a_dst[h * C + c];
    }
    als[t] = s1;
    ald[t] = s2;
}

// out[n,j] = bias[j]; mb = -inf; sb = 0
__global__ void init_kernel(float* __restrict__ mb, float* __restrict__ sb,
                            float* __restrict__ outb, const float* __restrict__ bias,
                            int N, int HC, int heads) {
    int t = blockIdx.x * blockDim.x + threadIdx.x;
    if (t < N * HC) outb[t] = bias[t % HC];
    if (t < N * heads) { mb[t] = -3.402823466e38f; sb[t] = 0.0f; }
}

__global__ void edge_max_kernel(const int* __restrict__ src, const int* __restrict__ dst,
                                const float* __restrict__ als, const float* __restrict__ ald,
                                float* __restrict__ mb, int E, int heads) {
    int e = blockIdx.x * blockDim.x + threadIdx.x;
    if (e >= E) return;
    int s = src[e], d = dst[e];
    for (int h = 0; h < heads; ++h) {
        float v = als[s * heads + h] + ald[d * heads + h];
        v = (v > 0.0f) ? v : LEAKY_SLOPE * v;
        atomicMaxF(&mb[d * heads + h], v);
    }
}

__global__ void edge_exp_kernel(const int* __restrict__ src, const int* __restrict__ dst,
                                const float* __restrict__ als, const float* __restrict__ ald,
                                const float* __restrict__ mb, float* __restrict__ sb,
                                float* __restrict__ exb, int E, int heads) {
    int e = blockIdx.x * blockDim.x + threadIdx.x;
    if (e >= E) return;
    int s = src[e], d = dst[e];
    for (int h = 0; h < heads; ++h) {
        float v = als[s * heads + h] + ald[d * heads + h];
        v = (v > 0.0f) ? v : LEAKY_SLOPE * v;
        float ex = expf(v - mb[d * heads + h]);
        exb[e * heads + h] = ex;
        atomicAdd(&sb[d * heads + h], ex);
    }
}

// out[dst,h,:] += h[src,h,:] * alpha(e,h)
__global__ void edge_agg_kernel(const int* __restrict__ src, const int* __restrict__ dst,
                                const float* __restrict__ Hf, const float* __restrict__ exb,
                                const float* __restrict__ sb, float* __restrict__ outb,
                                int E, int heads, int C) {
    int t = blockIdx.x * blockDim.x + threadIdx.x;
    if (t >= E * heads) return;
    int e = t / heads, h = t - e * heads;
    int s = src[e], d = dst[e];
    float alpha = exb[t] / (sb[d * heads + h] + 1e-16f);
    const float* hp = Hf + (size_t)s * heads * C + h * C;
    float* op = outb + (size_t)d * heads * C + h * C;
    for (int c = 0; c < C; ++c) atomicAdd(&op[c], hp[c] * alpha);
}

// act = elu(bn(out))   (bias already folded into out by init_kernel)
__global__ void post_kernel(const float* __restrict__ outb, const float* __restrict__ g,
                            const float* __restrict__ b, const float* __restrict__ mean,
                            const float* __restrict__ var, float* __restrict__ act,
                            int N, int HC) {
    int t = blockIdx.x * blockDim.x + threadIdx.x;
    if (t >= N * HC) return;
    int j = t % HC;
    float v = outb[t];
    v = (v - mean[j]) * rsqrtf(var[j] + BN_EPS_F) * g[j] + b[j];
    act[t] = (v > 0.0f) ? v : (expf(v) - 1.0f);
}

__global__ void cls_kernel(const float* __restrict__ act, const float* __restrict__ Wc,
                           const float* __restrict__ bc, float* __restrict__ out,
                           int N, int K, int NC) {
    int t = blockIdx.x * blockDim.x + threadIdx.x;
    if (t >= N * NC) return;
    int n = t / NC, j = t - n * NC;
    float acc = bc[j];
    const float* ap = act + (size_t)n * K;
    for (int k = 0; k < K; ++k) acc += ap[k] * Wc[k * NC + j];
    out[t] = acc;
}

extern "C" void kernel_launch(void* const* d_in, const int* in_sizes, int n_in,
                              void* d_out, int out_size, void* d_ws, size_t ws_size,
                              hipStream_t stream) {
    const int IN_CH = 128, C = 32, NCLS = 10;
    const float* x = (const float*)d_in[0];
    const int* ei = (const int*)d_in[1];
    const int E = in_sizes[1] / 2;
    const int N = in_sizes[0] / IN_CH;
    const int* src = ei;
    const int* dst = ei + E;

    const float* W[3]     = {(const float*)d_in[2],  (const float*)d_in[10], (const float*)d_in[18]};
    const float* a_s[3]   = {(const float*)d_in[3],  (const float*)d_in[11], (const float*)d_in[19]};
    const float* a_d[3]   = {(const float*)d_in[4],  (const float*)d_in[12], (const float*)d_in[20]};
    const float* bias[3]  = {(const float*)d_in[5],  (const float*)d_in[13], (const float*)d_in[21]};
    const float* gam[3]   = {(const float*)d_in[6],  (const float*)d_in[14], (const float*)d_in[22]};
    const float* bet[3]   = {(const float*)d_in[7],  (const float*)d_in[15], (const float*)d_in[23]};
    const float* rmean[3] = {(const float*)d_in[8],  (const float*)d_in[16], (const float*)d_in[24]};
    const float* rvar[3]  = {(const float*)d_in[9],  (const float*)d_in[17], (const float*)d_in[25]};
    const float* Wc = (const float*)d_in[26];
    const float* bc = (const float*)d_in[27];

    // workspace carve (floats): act | h | out | als | ald | m | s | ex
    float* bufAct = (float*)d_ws;
    float* bufH   = bufAct + (size_t)N * 128;
    float* bufOut = bufH   + (size_t)N * 128;
    float* als    = bufOut + (size_t)N * 128;
    float* ald    = als + (size_t)N * 4;
    float* mb     = ald + (size_t)N * 4;
    float* sb     = mb  + (size_t)N * 4;
    float* exb    = sb  + (size_t)N * 4;

    const float* actIn = x;
    int fin = IN_CH;
    for (int layer = 0; layer < 3; ++layer) {
        const int heads = (layer < 2) ? 4 : 1;
        const int HC = heads * C;   // fout
        // dense transform (WMMA f32)
        dim3 gblock(32, HC / 16);
        size_t shmem = (size_t)16 * (fin + 4) * sizeof(float);
        gat_gemm_wmma<<<(N + 15) / 16, gblock, shmem, stream>>>(actIn, W[layer], bufH, N, fin, HC);
        // attention logits per node/head
        int nh = N * heads;
        att_coef_kernel<<<(nh + 255) / 256, 256, 0, stream>>>(bufH, a_s[layer], a_d[layer],
                                                              als, ald, N, heads, C);
        // init accumulators
        int nhc = N * HC;
        init_kernel<<<(nhc + 255) / 256, 256, 0, stream>>>(mb, sb, bufOut, bias[layer], N, HC, heads);
        // segment softmax over dst + aggregation
        edge_max_kernel<<<(E + 255) / 256, 256, 0, stream>>>(src, dst, als, ald, mb, E, heads);
        edge_exp_kernel<<<(E + 255) / 256, 256, 0, stream>>>(src, dst, als, ald, mb, sb, exb, E, heads);
        int eh = E * heads;
        edge_agg_kernel<<<(eh + 255) / 256, 256, 0, stream>>>(src, dst, bufH, exb, sb, bufOut,
                                                              E, heads, C);
        // bias(+already folded) -> BN -> ELU
        post_kernel<<<(nhc + 255) / 256, 256, 0, stream>>>(bufOut, gam[layer], bet[layer],
                                                           rmean[layer], rvar[layer], bufAct, N, HC);
        actIn = bufAct;
        fin = HC;
    }
    // classifier: [N,32] @ [32,10] + bc
    int tot = N * NCLS;
    cls_kernel<<<(tot + 255) / 256, 256, 0, stream>>>(bufAct, Wc, bc, (float*)d_out, N, C, NCLS);
}